// Model_72129680769721
// MI455X (gfx1250) — compile-verified
//
#include <hip/hip_runtime.h>
#include <hip/hip_bf16.h>
#include <math.h>

typedef float v2f __attribute__((ext_vector_type(2)));
typedef float v4f __attribute__((ext_vector_type(4)));
typedef float v8f __attribute__((ext_vector_type(8)));

namespace {
constexpr int kB = 32, kT = 8, kC = 2048, kH = 24, kW = 8, kS = 6;
constexpr int kN = kB * kT;          // 256 rows (b*T+t)
constexpr int kD = kS * 256;         // 1536
constexpr int kHW = kH * kW;         // 192 floats per (n,c)
constexpr float kRS = 0.9999950000375f; // 1/sqrt(1+1e-5)
}

// ---------------------------------------------------------------------------
// Stage 1: masked spatial pooling. pooled[s][n][c] = mask[n] * mean of the 32
// contiguous floats feat[n,c, 4s..4s+3, 0..7]. Reads the 403MB tensor ONCE,
// with non-temporal policy so the stream does not evict the L2-resident
// weights / pooled working set used by every WMMA GEMM afterwards.
// ---------------------------------------------------------------------------
__global__ __launch_bounds__(256) void pool_kernel(const float* __restrict__ feat,
                                                   const float* __restrict__ mask,
                                                   float* __restrict__ pooled) {
  int idx = blockIdx.x * blockDim.x + threadIdx.x;
  const int total = kN * kC * kS;
  if (idx >= total) return;
  int s = idx % kS;
  int c = (idx / kS) % kC;
  int n = idx / (kS * kC);
  const v4f* p = reinterpret_cast<const v4f*>(feat + (size_t)(n * kC + c) * kHW + s * 32);
  float sum = 0.f;
#pragma unroll
  for (int i = 0; i < 8; ++i) {
    v4f v = __builtin_nontemporal_load(p + i);   // TH=NT: single-use stream
    sum += v.x + v.y + v.z + v.w;
  }
  pooled[((size_t)s * kN + n) * kC + c] = sum * (1.0f / 32.0f) * mask[n];
}

// ---------------------------------------------------------------------------
// Batched fp32 WMMA GEMM: C[m,n] = ep( sum_k A[m,k] * W[n,k] ), W row-major
// (N,K). One wave computes a 32x64 output tile: 2 (M) x 4 (N) 16x16 WMMA
// fragments = 8 independent v8f accumulators, V_WMMA_F32_16X16X4_F32.
// K is processed in chunks of 16: all 24 fragment loads (6 per K-step x 4
// steps) are issued first into distinct registers, then the 32-WMMA burst
// runs — so waits amortize over a whole chunk instead of per-load.
// __launch_bounds__(256,1) lets the compiler keep those loads in flight
// rather than recycling destination registers for occupancy.
// Fragment layouts (ISA 7.12.2, f32): A 16x4 and B 4x16 both store a
// contiguous K-pair per lane (lane<16: K={0,1}, lane>=16: K={2,3}), so both
// are plain v2f loads at column offset 2*kh.
// Epilogue: r = (acc + b1[n]) * (scale[n]*scaleMul) + b2[n]; optional ReLU.
// ---------------------------------------------------------------------------
__global__ __launch_bounds__(256, 1) void wmma_gemm_f32(
    const float* __restrict__ A, int lda, size_t strideA,
    const float* __restrict__ Bw, int ldb, size_t strideB,
    float* __restrict__ Cc, int ldc, size_t strideC,
    const float* __restrict__ bias1, int strideB1,
    const float* __restrict__ scale, int strideS,
    const float* __restrict__ bias2, int strideB2,
    int M, int N, int K, int relu, float scaleMul) {
  const int batch = blockIdx.z;
  A += (size_t)batch * strideA;
  Bw += (size_t)batch * strideB;
  Cc += (size_t)batch * strideC;
  if (bias1) bias1 += (size_t)batch * strideB1;
  if (scale) scale += (size_t)batch * strideS;
  if (bias2) bias2 += (size_t)batch * strideB2;

  const int wave = threadIdx.x >> 5;
  const int lane = threadIdx.x & 31;
  const int tilesN = N >> 6;                  // 64-wide N tiles
  const int tilesTotal = (M >> 5) * tilesN;   // 32-tall M tiles
  const int tile = blockIdx.x * (blockDim.x >> 5) + wave;
  if (tile >= tilesTotal) return;  // wave-uniform: EXEC stays all-ones for WMMA
  const int tm = tile / tilesN;
  const int tn = tile % tilesN;

  const int row = lane & 15;
  const int kh = lane >> 4;

  const float* pA0 = A + (size_t)(tm * 32 + row) * lda + 2 * kh;
  const float* pA1 = pA0 + (size_t)16 * lda;
  const float* pB0 = Bw + (size_t)(tn * 64 + row) * ldb + 2 * kh;
  const float* pB1 = pB0 + (size_t)16 * ldb;
  const float* pB2 = pB0 + (size_t)32 * ldb;
  const float* pB3 = pB0 + (size_t)48 * ldb;

  v8f c00 = {}, c01 = {}, c02 = {}, c03 = {};
  v8f c10 = {}, c11 = {}, c12 = {}, c13 = {};

#define WMMA(a, b, c) \
  c = __builtin_amdgcn_wmma_f32_16x16x4_f32(false, a, false, b, (short)0, c, false, false)

  for (int k = 0; k < K; k += 16) {           // 4 WMMA K-steps per chunk
    v2f fA0[4], fA1[4], fB0[4], fB1[4], fB2[4], fB3[4];
#pragma unroll
    for (int u = 0; u < 4; ++u) {             // 24 independent loads first
      const int kk = k + 4 * u;
      fA0[u] = *reinterpret_cast<const v2f*>(pA0 + kk);
      fA1[u] = *reinterpret_cast<const v2f*>(pA1 + kk);
      fB0[u] = *reinterpret_cast<const v2f*>(pB0 + kk);
      fB1[u] = *reinterpret_cast<const v2f*>(pB1 + kk);
      fB2[u] = *reinterpret_cast<const v2f*>(pB2 + kk);
      fB3[u] = *reinterpret_cast<const v2f*>(pB3 + kk);
    }
#pragma unroll
    for (int u = 0; u < 4; ++u) {             // 32-WMMA burst
      WMMA(fA0[u], fB0[u], c00); WMMA(fA0[u], fB1[u], c01);
      WMMA(fA0[u], fB2[u], c02); WMMA(fA0[u], fB3[u], c03);
      WMMA(fA1[u], fB0[u], c10); WMMA(fA1[u], fB1[u], c11);
      WMMA(fA1[u], fB2[u], c12); WMMA(fA1[u], fB3[u], c13);
    }
  }
#undef WMMA

  // Epilogue.  C/D layout: VGPR r -> M = r + 8*kh, N = lane&15.
  const int col = lane & 15;
  v8f accs[2][4] = {{c00, c01, c02, c03}, {c10, c11, c12, c13}};
#pragma unroll
  for (int jn = 0; jn < 4; ++jn) {
    const int nIdx = tn * 64 + jn * 16 + col;
    const float b1v = bias1 ? bias1[nIdx] : 0.f;
    const float sv = (scale ? scale[nIdx] : 1.f) * scaleMul;
    const float b2v = bias2 ? bias2[nIdx] : 0.f;
#pragma unroll
    for (int mi = 0; mi < 2; ++mi) {
#pragma unroll
      for (int r = 0; r < 8; ++r) {
        const int m = tm * 32 + mi * 16 + r + 8 * kh;
        float v = (accs[mi][jn][r] + b1v) * sv + b2v;
        if (relu) v = fmaxf(v, 0.f);
        Cc[(size_t)m * ldc + nIdx] = v;
      }
    }
  }
}

// h1[b,i,j,o] = relu(ha[b,j,o] + hb[b,i,o] + gb1[o])
__global__ __launch_bounds__(256) void h1_kernel(const float* __restrict__ ha,
                                                 const float* __restrict__ hb,
                                                 const float* __restrict__ gb1,
                                                 float* __restrict__ h1) {
  int idx = blockIdx.x * blockDim.x + threadIdx.x;
  const int total = kB * kT * kT * 768;
  if (idx >= total) return;
  int o = idx % 768;
  int j = (idx / 768) & 7;
  int i = (idx / (768 * 8)) & 7;
  int b = idx / (768 * 64);
  float v = ha[(size_t)(b * 8 + j) * 768 + o] + hb[(size_t)(b * 8 + i) * 768 + o] + gb1[o];
  h1[idx] = fmaxf(v, 0.f);
}

// gout[n,p] = sum_j h4[n,j,p]   (n = b*T+i)
__global__ __launch_bounds__(256) void gout_kernel(const float* __restrict__ h4,
                                                   float* __restrict__ gout) {
  int idx = blockIdx.x * blockDim.x + threadIdx.x;
  const int total = kN * 512;
  if (idx >= total) return;
  int p = idx % 512;
  int n = idx / 512;
  float s = 0.f;
#pragma unroll
  for (int j = 0; j < 8; ++j) s += h4[(size_t)(n * 8 + j) * 512 + p];
  gout[idx] = s;
}

// Self-attention + co-attention + normalization (tiny: one block, 256 threads)
__global__ __launch_bounds__(256) void att_kernel(
    const float* __restrict__ sc, const float* __restrict__ tconv_w,
    const float* __restrict__ tconv_b, const float* __restrict__ tbn_g,
    const float* __restrict__ tbn_b, const float* __restrict__ f1,
    const float* __restrict__ fw2, const float* __restrict__ fb2,
    const float* __restrict__ mask, float* __restrict__ att) {
  const int n = threadIdx.x;  // 0..255
  float tv = 0.f;
  for (int d = 0; d < kD; ++d) tv += sc[(size_t)n * kD + d] * tconv_w[d];
  tv += tconv_b[0];
  tv = tv * (tbn_g[0] * kRS) + tbn_b[0];
  tv = fmaxf(tv, 0.f);
  tv = (1.f / (1.f + __expf(-tv))) * mask[n];

  float fv = 0.f;
  for (int d = 0; d < 128; ++d) fv += f1[(size_t)n * 128 + d] * fw2[d];
  fv += fb2[0];
  float cv = (1.f / (1.f + __expf(-fv))) * mask[n];

  __shared__ float sta[kN];
  __shared__ float sca[kN];
  sta[n] = tv;
  sca[n] = cv;
  __syncthreads();
  const int b0 = n & ~7;
  float st = 0.f, scm = 0.f;
#pragma unroll
  for (int j = 0; j < 8; ++j) { st += sta[b0 + j]; scm += sca[b0 + j]; }
  att[n] = 0.5f * (tv / st + cv / scm);
}

// aggT[s][b][c] = sum_t att[b*T+t] * pooled[s][b*T+t][c]  (pooling is linear,
// so the 403MB feat_map is never touched again)
__global__ __launch_bounds__(256) void agg_kernel(const float* __restrict__ pooled,
                                                  const float* __restrict__ att,
                                                  float* __restrict__ aggT) {
  int idx = blockIdx.x * blockDim.x + threadIdx.x;
  const int total = kS * kB * kC;
  if (idx >= total) return;
  int c = idx % kC;
  int b = (idx / kC) % kB;
  int s = idx / (kC * kB);
  float a = 0.f;
#pragma unroll
  for (int t = 0; t < 8; ++t)
    a += att[b * 8 + t] * pooled[((size_t)s * kN + b * 8 + t) * kC + c];
  aggT[((size_t)s * kB + b) * kC + c] = a;
}

extern "C" void kernel_launch(void* const* d_in, const int* in_sizes, int n_in,
                              void* d_out, int out_size, void* d_ws, size_t ws_size,
                              hipStream_t stream) {
  const float* feat    = (const float*)d_in[0];
  const float* mask    = (const float*)d_in[1];
  const float* lconv_w = (const float*)d_in[2];   // (S,256,2048)
  const float* lconv_b = (const float*)d_in[3];   // (S,256)
  const float* lbn_g   = (const float*)d_in[4];
  const float* lbn_b   = (const float*)d_in[5];
  const float* tconv_w = (const float*)d_in[6];
  const float* tconv_b = (const float*)d_in[7];
  const float* tbn_g   = (const float*)d_in[8];
  const float* tbn_b   = (const float*)d_in[9];
  const float* gw1     = (const float*)d_in[10];  // (768, 3072)
  const float* gb1     = (const float*)d_in[11];
  const float* gw2     = (const float*)d_in[12];  // (512, 768)
  const float* gb2     = (const float*)d_in[13];
  const float* gw3     = (const float*)d_in[14];  // (512, 512)
  const float* gb3     = (const float*)d_in[15];
  const float* gw4     = (const float*)d_in[16];  // (512, 512)
  const float* gb4     = (const float*)d_in[17];
  const float* fw1     = (const float*)d_in[18];  // (128, 512)
  const float* fb1     = (const float*)d_in[19];
  const float* fw2     = (const float*)d_in[20];  // (1, 128)
  const float* fb2     = (const float*)d_in[21];
  float* out = (float*)d_out;                     // (32, 6, 256)

  // workspace layout (floats)
  float* ws = (float*)d_ws;
  size_t off = 0;
  float* pooled = ws + off; off += (size_t)kS * kN * kC;   // 3,145,728
  float* sc     = ws + off; off += (size_t)kN * kD;        //   393,216
  float* ha     = ws + off; off += (size_t)kN * 768;       //   196,608
  float* hb     = ws + off; off += (size_t)kN * 768;       //   196,608
  float* h1     = ws + off; off += (size_t)kN * 8 * 768;   // 1,572,864
  float* h2     = ws + off; off += (size_t)kN * 8 * 512;   // 1,048,576
  float* h3     = ws + off; off += (size_t)kN * 8 * 512;
  float* h4     = ws + off; off += (size_t)kN * 8 * 512;
  float* gout   = ws + off; off += (size_t)kN * 512;
  float* f1     = ws + off; off += (size_t)kN * 128;
  float* tatt   = ws + off; off += kN;
  float* aggT   = ws + off; off += (size_t)kS * kB * kC;

  auto blocksFor = [](int M, int N) { return ((M / 32) * (N / 64) + 7) / 8; };

  // 1) pooled (single NT pass over feat_map)
  pool_kernel<<<(kN * kC * kS) / 256, 256, 0, stream>>>(feat, mask, pooled);

  // 2) sc[n, s*256+o] = (pooled_s . lconv_w_s^T + lconv_b)*lbn_g*RS + lbn_b
  wmma_gemm_f32<<<dim3(blocksFor(kN, 256), 1, kS), 256, 0, stream>>>(
      pooled, kC, (size_t)kN * kC, lconv_w, kC, (size_t)256 * kC,
      sc, kD, 256, lconv_b, 256, lbn_g, 256, lbn_b, 256,
      kN, 256, kC, 0, kRS);

  // 3) ha = sc . wa^T, hb = sc . wb^T  (batch=2 slicing gw1 with ldb=3072)
  wmma_gemm_f32<<<dim3(blocksFor(kN, 768), 1, 2), 256, 0, stream>>>(
      sc, kD, 0, gw1, 2 * kD, (size_t)kD,
      ha, 768, (size_t)kN * 768, nullptr, 0, nullptr, 0, nullptr, 0,
      kN, 768, kD, 0, 1.0f);

  // 4) h1 = relu(ha[b,j]+hb[b,i]+gb1)
  h1_kernel<<<(kB * 64 * 768) / 256, 256, 0, stream>>>(ha, hb, gb1, h1);

  // 5-7) MLP chain with fused bias+ReLU
  wmma_gemm_f32<<<dim3(blocksFor(kN * 8, 512), 1, 1), 256, 0, stream>>>(
      h1, 768, 0, gw2, 768, 0, h2, 512, 0,
      nullptr, 0, nullptr, 0, gb2, 0, kN * 8, 512, 768, 1, 1.0f);
  wmma_gemm_f32<<<dim3(blocksFor(kN * 8, 512), 1, 1), 256, 0, stream>>>(
      h2, 512, 0, gw3, 512, 0, h3, 512, 0,
      nullptr, 0, nullptr, 0, gb3, 0, kN * 8, 512, 512, 1, 1.0f);
  wmma_gemm_f32<<<dim3(blocksFor(kN * 8, 512), 1, 1), 256, 0, stream>>>(
      h3, 512, 0, gw4, 512, 0, h4, 512, 0,
      nullptr, 0, nullptr, 0, gb4, 0, kN * 8, 512, 512, 1, 1.0f);

  // 8) gout = sum_j h4
  gout_kernel<<<(kN * 512) / 256, 256, 0, stream>>>(h4, gout);

  // 9) f1 = relu(gout . fw1^T + fb1)
  wmma_gemm_f32<<<dim3(blocksFor(kN, 128), 1, 1), 256, 0, stream>>>(
      gout, 512, 0, fw1, 512, 0, f1, 128, 0,
      nullptr, 0, nullptr, 0, fb1, 0, kN, 128, 512, 1, 1.0f);

  // 10) attention weights
  att_kernel<<<1, 256, 0, stream>>>(sc, tconv_w, tconv_b, tbn_g, tbn_b,
                                    f1, fw2, fb2, mask, tatt);

  // 11) aggT = att-weighted pooled (feat_map NOT re-read)
  agg_kernel<<<(kS * kB * kC) / 256, 256, 0, stream>>>(pooled, tatt, aggT);

  // 12) local = spatial conv of agg -> d_out (32,6,256)
  wmma_gemm_f32<<<dim3(blocksFor(kB, 256), 1, kS), 256, 0, stream>>>(
      aggT, kC, (size_t)kB * kC, lconv_w, kC, (size_t)256 * kC,
      out, kD, 256, lconv_b, 256, lbn_g, 256, lbn_b, 256,
      kB, 256, kC, 0, kRS);
}